// SinkhornDistance_19421842113220
// MI455X (gfx1250) — compile-verified
//
#include <hip/hip_runtime.h>
#include <math.h>

#define PP    2048
#define BB    4
#define DD    64
#define EPSF  0.1f
#define IEPS  10.0f
#define THRF  0.1f
#define NITER 50

typedef float v2f __attribute__((ext_vector_type(2)));
typedef float v8f __attribute__((ext_vector_type(8)));

__device__ __forceinline__ float warpMax32(float v) {
#pragma unroll
    for (int off = 16; off > 0; off >>= 1)
        v = fmaxf(v, __shfl_xor(v, off, 32));
    return v;
}
__device__ __forceinline__ float warpSum32(float v) {
#pragma unroll
    for (int off = 16; off > 0; off >>= 1)
        v += __shfl_xor(v, off, 32);
    return v;
}

// ---- init u = v = 0, done = 0 ------------------------------------------
__global__ void sk_init(float* __restrict__ u, float* __restrict__ v,
                        int* __restrict__ done) {
    int i = blockIdx.x * blockDim.x + threadIdx.x;   // 8192 threads
    u[i] = 0.0f;
    v[i] = 0.0f;
    if (i == 0) *done = 0;
}

// ---- squared norms of x and y rows (one wave per row) ------------------
__global__ void sk_norms(const float* __restrict__ x, const float* __restrict__ y,
                         float* __restrict__ xn, float* __restrict__ yn) {
    int wid  = blockIdx.x * 8 + (threadIdx.x >> 5);  // 16384 waves total
    int lane = threadIdx.x & 31;
    const float* src;
    float* dst;
    int row;
    if (wid < BB * PP) { src = x; dst = xn; row = wid; }
    else               { src = y; dst = yn; row = wid - BB * PP; }
    const float* p = src + (size_t)row * DD;
    float a0 = p[lane], a1 = p[lane + 32];
    float s = warpSum32(a0 * a0 + a1 * a1);
    if (lane == 0) dst[row] = s;
}

// ---- C = |x|^2 + |y|^2 - 2 x.y  via V_WMMA_F32_16X16X4_F32 -------------
// One wave computes one 16x16 tile of C; K=64 -> 16 WMMA steps.
__global__ void sk_cost(const float* __restrict__ x, const float* __restrict__ y,
                        const float* __restrict__ xn, const float* __restrict__ yn,
                        float* __restrict__ C) {
    int wid  = blockIdx.x * 8 + (threadIdx.x >> 5);  // 65536 tiles
    int lane = threadIdx.x & 31;
    int b    = wid >> 14;             // /(128*128)
    int rem  = wid & 16383;
    int i0   = (rem >> 7) << 4;       // row tile * 16
    int j0   = (rem & 127) << 4;      // col tile * 16
    int lm   = lane & 15;
    int half = lane >> 4;

    // A fragment: lane holds X[i0+lm, k0 + 2*half .. +1]
    // B fragment: lane holds Y[j0+lm, k0 + 2*half .. +1]  (B = Y^T, 4x16)
    const float* xa = x + (size_t)(b * PP + i0 + lm) * DD + 2 * half;
    const float* yb = y + (size_t)(b * PP + j0 + lm) * DD + 2 * half;

    v8f acc = {0.f, 0.f, 0.f, 0.f, 0.f, 0.f, 0.f, 0.f};
#pragma unroll
    for (int kk = 0; kk < 16; ++kk) {
        v2f a = *(const v2f*)(xa + kk * 4);
        v2f bm = *(const v2f*)(yb + kk * 4);
        acc = __builtin_amdgcn_wmma_f32_16x16x4_f32(
            false, a, false, bm, (short)0, acc, false, false);
    }

    float ynv = yn[b * PP + j0 + lm];     // N = lm for both lane halves
#pragma unroll
    for (int r = 0; r < 8; ++r) {
        int m = r + 8 * half;             // D layout: VGPR r -> M = r + 8*half
        float val = xn[b * PP + i0 + m] + ynv - 2.0f * acc[r];
        C[(size_t)(b * PP + i0 + m) * PP + (j0 + lm)] = val;
    }
}

// ---- row pass: u_i = eps*(log_mu - LSE_j((v_j - C_ij)/eps)), err accum --
// 1024 blocks, 8 waves/block, one wave per row; row cached in 64 regs/lane.
__global__ void sk_row(const float* __restrict__ C, float* __restrict__ u,
                       const float* __restrict__ v, float* __restrict__ errP,
                       const int* __restrict__ done, float log_mu) {
    if (*done) return;
    __shared__ float vs[PP];
    __shared__ float duw[8];

    int b       = blockIdx.x >> 8;                 // 256 blocks per batch
    int rowBase = (blockIdx.x & 255) << 3;         // 8 rows per block
    int wave    = threadIdx.x >> 5;
    int lane    = threadIdx.x & 31;

    for (int idx = threadIdx.x; idx < PP; idx += 256)
        vs[idx] = v[b * PP + idx];
    __syncthreads();

    int i = rowBase + wave;
    const float* Crow = C + (size_t)(b * PP + i) * PP;

    float t[64];
    float mx = -INFINITY;
#pragma unroll
    for (int jj = 0; jj < 64; ++jj) {
        int j = jj * 32 + lane;
        float tv = (vs[j] - Crow[j]) * IEPS;
        t[jj] = tv;
        mx = fmaxf(mx, tv);
    }
    mx = warpMax32(mx);
    float s = 0.0f;
#pragma unroll
    for (int jj = 0; jj < 64; ++jj)
        s += __expf(t[jj] - mx);
    s = warpSum32(s);

    if (lane == 0) {
        float K    = mx + __logf(s);
        float unew = EPSF * (log_mu - K);
        float uold = u[b * PP + i];
        u[b * PP + i] = unew;
        duw[wave] = fabsf(unew - uold);
    }
    __syncthreads();
    if (threadIdx.x == 0) {
        float e = 0.0f;
#pragma unroll
        for (int w = 0; w < 8; ++w) e += duw[w];
        errP[blockIdx.x] = e;
    }
}

// ---- col pass: v_j = eps*(log_nu - LSE_i((u_i - C_ij)/eps)) ------------
// 256 blocks; block = 32 columns x 8 row-groups; coalesced 128B row reads.
__global__ void sk_col(const float* __restrict__ C, const float* __restrict__ u,
                       float* __restrict__ v, const int* __restrict__ done,
                       float log_nu) {
    if (*done) return;
    __shared__ float us[PP];
    __shared__ float red[8][32];

    int b  = blockIdx.x >> 6;               // 64 col-tiles per batch
    int j0 = (blockIdx.x & 63) << 5;        // 32 columns per block
    int c  = threadIdx.x & 31;
    int r  = threadIdx.x >> 5;
    int j  = j0 + c;

    for (int idx = threadIdx.x; idx < PP; idx += 256)
        us[idx] = u[b * PP + idx];
    __syncthreads();

    const float* Cb = C + (size_t)b * PP * PP;

    float pmax = -INFINITY;
    for (int i = r; i < PP; i += 8)
        pmax = fmaxf(pmax, (us[i] - Cb[(size_t)i * PP + j]) * IEPS);
    red[r][c] = pmax;
    __syncthreads();
    if (r == 0) {
        float m = red[0][c];
#pragma unroll
        for (int rr = 1; rr < 8; ++rr) m = fmaxf(m, red[rr][c]);
        red[0][c] = m;
    }
    __syncthreads();
    float cm = red[0][c];
    __syncthreads();

    float ps = 0.0f;
    for (int i = r; i < PP; i += 8)
        ps += __expf((us[i] - Cb[(size_t)i * PP + j]) * IEPS - cm);
    red[r][c] = ps;
    __syncthreads();
    if (r == 0) {
        float s = red[0][c];
#pragma unroll
        for (int rr = 1; rr < 8; ++rr) s += red[rr][c];
        float K = cm + __logf(s);
        v[b * PP + j] = EPSF * (log_nu - K);
    }
}

// ---- err = mean_b sum_i |du| ; set done if err < thresh ----------------
__global__ void sk_flag(const float* __restrict__ errP, int* __restrict__ done) {
    if (*done) return;
    __shared__ float sd[256];
    int t = threadIdx.x;
    sd[t] = errP[t] + errP[t + 256] + errP[t + 512] + errP[t + 768];
    __syncthreads();
    for (int off = 128; off > 0; off >>= 1) {
        if (t < off) sd[t] += sd[t + off];
        __syncthreads();
    }
    if (t == 0 && sd[0] * (1.0f / BB) < THRF) *done = 1;
}

// ---- partial sums of pi * C --------------------------------------------
__global__ void sk_pi(const float* __restrict__ C, const float* __restrict__ u,
                      const float* __restrict__ v, float* __restrict__ costP) {
    __shared__ float sd[256];
    const size_t N = (size_t)BB * PP * PP;
    float acc = 0.0f;
    for (size_t idx = (size_t)blockIdx.x * 256 + threadIdx.x; idx < N;
         idx += (size_t)4096 * 256) {
        size_t b   = idx >> 22;               // /(P*P)
        size_t rem = idx & ((size_t)PP * PP - 1);
        int i = (int)(rem >> 11);
        int j = (int)(rem & 2047);
        float Cv = C[idx];
        acc += __expf((u[b * PP + i] + v[b * PP + j] - Cv) * IEPS) * Cv;
    }
    sd[threadIdx.x] = acc;
    __syncthreads();
    for (int off = 128; off > 0; off >>= 1) {
        if (threadIdx.x < off) sd[threadIdx.x] += sd[threadIdx.x + off];
        __syncthreads();
    }
    if (threadIdx.x == 0) costP[blockIdx.x] = sd[0];
}

__global__ void sk_final(const float* __restrict__ costP, float* __restrict__ out) {
    __shared__ float sd[256];
    float s = 0.0f;
    for (int k = threadIdx.x; k < 4096; k += 256) s += costP[k];
    sd[threadIdx.x] = s;
    __syncthreads();
    for (int off = 128; off > 0; off >>= 1) {
        if (threadIdx.x < off) sd[threadIdx.x] += sd[threadIdx.x + off];
        __syncthreads();
    }
    if (threadIdx.x == 0) out[0] = sd[0] * (1.0f / BB);
}

extern "C" void kernel_launch(void* const* d_in, const int* in_sizes, int n_in,
                              void* d_out, int out_size, void* d_ws, size_t ws_size,
                              hipStream_t stream) {
    const float* x = (const float*)d_in[0];
    const float* y = (const float*)d_in[1];
    float* out = (float*)d_out;

    float* ws    = (float*)d_ws;
    float* C     = ws;                         // 16,777,216 floats (64 MiB)
    float* xn    = C + (size_t)BB * PP * PP;   // 8192
    float* yn    = xn + BB * PP;               // 8192
    float* u     = yn + BB * PP;               // 8192
    float* v     = u + BB * PP;                // 8192
    float* errP  = v + BB * PP;                // 1024
    float* costP = errP + 1024;                // 4096
    int*   done  = (int*)(costP + 4096);

    const float log_mu = logf(1.0f / (float)PP + 1e-8f);
    const float log_nu = log_mu;

    sk_init <<<32,   256, 0, stream>>>(u, v, done);
    sk_norms<<<2048, 256, 0, stream>>>(x, y, xn, yn);
    sk_cost <<<8192, 256, 0, stream>>>(x, y, xn, yn, C);

    for (int it = 0; it < NITER; ++it) {
        sk_row <<<1024, 256, 0, stream>>>(C, u, v, errP, done, log_mu);
        sk_col <<<256,  256, 0, stream>>>(C, u, v, done, log_nu);
        sk_flag<<<1,    256, 0, stream>>>(errP, done);
    }

    sk_pi   <<<4096, 256, 0, stream>>>(C, u, v, costP);
    sk_final<<<1,    256, 0, stream>>>(costP, out);
}